// KalmanForecaster_62646392980130
// MI455X (gfx1250) — compile-verified
//
#include <hip/hip_runtime.h>
#include <stdint.h>

#define B_DIM 16384
#define L_DIM 512
#define H_DIM 128
#define ROWS 128        // batch rows per block == blockDim.x
#define TC 16           // timesteps per DMA chunk
#define LDS_STRIDE 17   // 16 data floats + 1 pad float (TDM pad_enable)

typedef unsigned int u32x4 __attribute__((ext_vector_type(4)));
typedef int i32x8 __attribute__((ext_vector_type(8)));
typedef int i32x4 __attribute__((ext_vector_type(4)));

__device__ __forceinline__ unsigned lds_off_u32(const void* p) {
  // generic pointer to LDS: low 32 bits hold the LDS byte offset
  return (unsigned)(uintptr_t)p;
}

// Issue one 2D tile load (ROWS x TC floats, row stride = stride_elems) via the
// Tensor Data Mover.  LDS layout: row r at lds_off + r*17*4 (pad after every
// 16 DWORDs -> conflict-free lane access, stride 17 is coprime with 64 banks).
__device__ __forceinline__ void tdm_load_tile(const float* gsrc, unsigned lds_off,
                                              unsigned stride_elems) {
  unsigned long long ga = (unsigned long long)(uintptr_t)gsrc;
  u32x4 g0;
  g0.x = 1u;                                    // count = 1 valid descriptor
  g0.y = lds_off;                               // lds_addr (bytes)
  g0.z = (unsigned)(ga & 0xffffffffu);          // global_addr[31:0]
  g0.w = (unsigned)((ga >> 32) & 0x01ffffffu)   // global_addr[56:32]
         | (2u << 30);                          // type = 2 ("image")
  i32x8 g1;
  g1[0] = (int)0x00D20000u;  // data_size=4B(2<<16) | pad_enable(1<<20) | pad_interval=16DW(3<<22) | pad_amount=1DW(0<<25)
  g1[1] = (int)(((unsigned)TC)   << 16);  // tensor_dim0 = TC (tile never OOB)
  g1[2] = (int)(((unsigned)ROWS) << 16);  // tensor_dim1 = ROWS
  g1[3] = (int)(((unsigned)TC)   << 16);  // tile_dim0 = TC
  g1[4] = (int)ROWS;                      // tile_dim1 = ROWS, tile_dim2 = 0
  g1[5] = (int)stride_elems;              // tensor_dim0_stride[31:0]
  g1[6] = 0;                              // stride hi / dim1_stride lo
  g1[7] = 0;
  i32x4 z4 = (i32x4){0, 0, 0, 0};              // groups 2/3 unused (2-D tensor)
  i32x8 z8 = (i32x8){0, 0, 0, 0, 0, 0, 0, 0};  // trailing group (6-arg form)
  __builtin_amdgcn_tensor_load_to_lds(g0, g1, z4, z4, z8, 0);
}

__device__ __forceinline__ float softplusf(float v) { return log1pf(__expf(v)); }

struct KFParams {
  float alpha, vcsq, kap, delt, qx, qu, R;
};

__device__ __forceinline__ void kf_predict(float& x, float& u,
                                           float& p00, float& p01, float& p11,
                                           float v, float dt, float gc,
                                           const KFParams& P) {
  float rho  = __expf(-P.alpha * dt);
  float rel  = v - u;
  float arel = fabsf(rel);
  float forcing = fmaxf(fmaf(v, v, -P.vcsq), 0.0f);
  float x_pred = fmaf(u, dt, x);
  float u_pred = rho * u - P.kap * x * dt + gc * forcing * dt
               + P.delt * rel * arel * dt;
  float f10 = -P.kap * dt;
  float f11 = rho - 2.0f * P.delt * arel * dt;
  // P_pred = F P F^T (+ process noise), symmetric 3-scalar form
  float t0   = fmaf(dt, p01, p00);           // (F P)[0,0]
  float t1   = fmaf(dt, p11, p01);           // (F P)[0,1]
  float np00 = fmaf(dt, t1, t0) + P.qx * dt;
  float np01 = f10 * t0 + f11 * t1;
  float s0   = f10 * p00 + f11 * p01;
  float s1   = f10 * p01 + f11 * p11;
  float np11 = f10 * s0 + f11 * s1 + P.qu * dt;
  x = x_pred; u = u_pred; p00 = np00; p01 = np01; p11 = np11;
}

__device__ __forceinline__ void kf_update(float& x, float& u,
                                          float& p00, float& p01, float& p11,
                                          float y, float R) {
  float S    = p00 + R;
  float invS = 1.0f / S;
  float K0   = p00 * invS;
  float K1   = p01 * invS;
  float innov = y - x;
  float om   = 1.0f - K0;
  x = fmaf(K0, innov, x);
  u = fmaf(K1, innov, u);
  // P_upd = (I-KH) P (I-KH)^T + R K K^T, symmetric
  float m00 = om * p00;
  float m01 = om * p01;
  float m10 = p01 - K1 * p00;
  float m11 = p11 - K1 * p01;
  float rp00 = om * m00 + R * K0 * K0;
  float rp01 = m01 - K1 * m00 + R * K0 * K1;
  float rp11 = m11 - K1 * m10 + R * K1 * K1;
  p00 = rp00; p01 = rp01; p11 = rp11;
}

__global__ __launch_bounds__(ROWS) void kalman_forecaster_kernel(
    const float* __restrict__ v_hist, const float* __restrict__ dt_hist,
    const float* __restrict__ x_obs,  const float* __restrict__ v_fut,
    const float* __restrict__ dt_fut,
    const float* pA, const float* pC, const float* pVC, const float* pK,
    const float* pG, const float* pD, const float* pQX, const float* pQU,
    const float* pR, const float* pP0X, const float* pP0U,
    float* __restrict__ out)
{
  __shared__ float ldsV[2][ROWS * LDS_STRIDE];
  __shared__ float ldsD[2][ROWS * LDS_STRIDE];
  __shared__ float ldsX[2][ROWS * LDS_STRIDE];

  const int tid = threadIdx.x;
  const int b0  = blockIdx.x * ROWS;
  const int b   = b0 + tid;

  KFParams prm;
  prm.alpha   = softplusf(*pA);
  float cpar  = *pC;
  float vc    = softplusf(*pVC);
  prm.vcsq    = vc * vc;
  prm.kap     = softplusf(*pK);
  float gamma = softplusf(*pG);
  prm.delt    = softplusf(*pD);
  prm.qx      = __expf(*pQX);
  prm.qu      = __expf(*pQU);
  prm.R       = __expf(*pR);

  float x   = x_obs[(size_t)b * L_DIM];  // s0 = [x_obs[:,0], 0]
  float u   = 0.0f;
  float p00 = __expf(*pP0X);
  float p01 = 0.0f;
  float p11 = __expf(*pP0U);

  const int NC1 = L_DIM / TC;  // 32 history chunks
  const int NC2 = H_DIM / TC;  // 8 future chunks

  // Prologue: DMA history chunk 0 (wave 0 issues; EXEC ignored by TDM)
  if (tid < 32) {
    const size_t rb = (size_t)b0 * L_DIM;
    tdm_load_tile(v_hist  + rb, lds_off_u32(&ldsV[0][0]), L_DIM);
    tdm_load_tile(dt_hist + rb, lds_off_u32(&ldsD[0][0]), L_DIM);
    tdm_load_tile(x_obs   + rb, lds_off_u32(&ldsX[0][0]), L_DIM);
  }

  // ---- Phase 1: filter over history (steps s = 0..510) ----
  // step s uses v[s], dt[s+1], x_obs[s+1]; within chunk c, k indexes cols
  // c*TC..c*TC+15 and step s = c*TC + k - 1 (v comes from k-1, carried across
  // chunk boundaries in a register).
  float v_carry = 0.0f;
  for (int c = 0; c < NC1; ++c) {
    const int bi = c & 1;
    if (tid < 32) __builtin_amdgcn_s_wait_tensorcnt(0);
    __syncthreads();  // chunk c resident for all waves; prev buffer free
    if (tid < 32) {
      if (c + 1 < NC1) {
        const size_t rb = (size_t)b0 * L_DIM + (size_t)(c + 1) * TC;
        tdm_load_tile(v_hist  + rb, lds_off_u32(&ldsV[bi ^ 1][0]), L_DIM);
        tdm_load_tile(dt_hist + rb, lds_off_u32(&ldsD[bi ^ 1][0]), L_DIM);
        tdm_load_tile(x_obs   + rb, lds_off_u32(&ldsX[bi ^ 1][0]), L_DIM);
      } else {  // overlap: prefetch first future chunk
        const size_t rb2 = (size_t)b0 * H_DIM;
        tdm_load_tile(v_fut  + rb2, lds_off_u32(&ldsV[bi ^ 1][0]), H_DIM);
        tdm_load_tile(dt_fut + rb2, lds_off_u32(&ldsD[bi ^ 1][0]), H_DIM);
      }
    }
    const float* Vt = &ldsV[bi][tid * LDS_STRIDE];
    const float* Dt = &ldsD[bi][tid * LDS_STRIDE];
    const float* Xt = &ldsX[bi][tid * LDS_STRIDE];
#pragma unroll
    for (int k = 0; k < TC; ++k) {
      if (c == 0 && k == 0) continue;  // no step s = -1
      float v  = (k == 0) ? v_carry : Vt[k - 1];
      float dt = fmaxf(Dt[k], 1e-6f);
      float y  = Xt[k];
      kf_predict(x, u, p00, p01, p11, v, dt, /*g*c = 1*c*/ cpar, prm);
      kf_update(x, u, p00, p01, p11, y, prm.R);
    }
    v_carry = Vt[TC - 1];
  }

  // ---- Phase 2: predict-only over future horizon ----
  const size_t ob  = (size_t)b * H_DIM;
  float* out_xp = out;
  float* out_xv = out + (size_t)B_DIM * H_DIM;
  float* out_ue = out + 2 * (size_t)B_DIM * H_DIM;
  const float gc2 = gamma * cpar;

  for (int p = 0; p < NC2; ++p) {
    const int bi = p & 1;  // chunk0 landed in buffer 0 (issued at c = NC1-1, bi^1 = 0)
    if (tid < 32) __builtin_amdgcn_s_wait_tensorcnt(0);
    __syncthreads();
    if (tid < 32 && p + 1 < NC2) {
      const size_t rb2 = (size_t)b0 * H_DIM + (size_t)(p + 1) * TC;
      tdm_load_tile(v_fut  + rb2, lds_off_u32(&ldsV[bi ^ 1][0]), H_DIM);
      tdm_load_tile(dt_fut + rb2, lds_off_u32(&ldsD[bi ^ 1][0]), H_DIM);
    }
    const float* Vt = &ldsV[bi][tid * LDS_STRIDE];
    const float* Dt = &ldsD[bi][tid * LDS_STRIDE];
    float axp[4], axv[4], aue[4];
#pragma unroll
    for (int k = 0; k < TC; ++k) {
      float v  = Vt[k];
      float dt = fmaxf(Dt[k], 1e-6f);
      kf_predict(x, u, p00, p01, p11, v, dt, gc2, prm);
      axp[k & 3] = x;
      axv[k & 3] = p00;
      aue[k & 3] = u;
      if ((k & 3) == 3) {  // 16-byte stores; L2 write-combines the row
        const int h0 = p * TC + (k - 3);
        *(float4*)(out_xp + ob + h0) = make_float4(axp[0], axp[1], axp[2], axp[3]);
        *(float4*)(out_xv + ob + h0) = make_float4(axv[0], axv[1], axv[2], axv[3]);
        *(float4*)(out_ue + ob + h0) = make_float4(aue[0], aue[1], aue[2], aue[3]);
      }
    }
  }
}

extern "C" void kernel_launch(void* const* d_in, const int* in_sizes, int n_in,
                              void* d_out, int out_size, void* d_ws, size_t ws_size,
                              hipStream_t stream) {
  (void)in_sizes; (void)n_in; (void)out_size; (void)d_ws; (void)ws_size;
  const float* v_hist  = (const float*)d_in[0];
  const float* dt_hist = (const float*)d_in[1];
  const float* x_obs   = (const float*)d_in[2];
  const float* v_fut   = (const float*)d_in[3];
  const float* dt_fut  = (const float*)d_in[4];
  const float* pA   = (const float*)d_in[5];
  const float* pC   = (const float*)d_in[6];
  const float* pVC  = (const float*)d_in[7];
  const float* pK   = (const float*)d_in[8];
  const float* pG   = (const float*)d_in[9];
  const float* pD   = (const float*)d_in[10];
  const float* pQX  = (const float*)d_in[11];
  const float* pQU  = (const float*)d_in[12];
  const float* pR   = (const float*)d_in[13];
  const float* pP0X = (const float*)d_in[14];
  const float* pP0U = (const float*)d_in[15];

  dim3 grid(B_DIM / ROWS);  // 128 blocks
  dim3 block(ROWS);         // 128 threads = 4 waves
  kalman_forecaster_kernel<<<grid, block, 0, stream>>>(
      v_hist, dt_hist, x_obs, v_fut, dt_fut,
      pA, pC, pVC, pK, pG, pD, pQX, pQU, pR, pP0X, pP0U,
      (float*)d_out);
}